// FFTConv_11622181503191
// MI455X (gfx1250) — compile-verified
//
#include <hip/hip_runtime.h>

typedef __attribute__((ext_vector_type(2))) float v2f;
typedef __attribute__((ext_vector_type(8))) float v8f;
typedef __attribute__((ext_vector_type(2))) int   v2i;

#define BATCH 8
#define CIN   32
#define COUT  32
#define HW    (128 * 128)
#define PIX   16            // pixels per workgroup: 16 complex = one 128B line per row
#define CT    4             // c-chunk size (one WMMA K-step per chunk)
#define NCHUNK (CIN / CT)   // 8

// ---- LDS layout (dword units) ----
// z buffer:    [k:4][b-row:8][pix:16]{re,im}, row stride 34, k stride KZ
// filt buffer: [k:4][p:32]  [pix:16]{re,im}, row stride 34, k stride KS
// KZ=272 (== 8*34, and ==16 mod 32) and KS=1104 (==16 mod 32) make the two
// lane-halves of each WMMA operand read land on disjoint LDS banks.
#define RS     34
#define KZ     272
#define ZSZ    (CT * KZ)          // 1088
#define KS     1104
#define FSZ    (CT * KS)          // 4416
#define BUFSZ  (ZSZ + FSZ)        // 5504
#define LDS_DW (2 * BUFSZ)        // 11008 dwords = 43 KB (double buffered)
// epilogue overlay (reuses LDS after final barrier): out planes [pix][b:8][p:32]
#define OP     260
#define OLR    0
#define OLI    (PIX * OP)         // 4160 .. 8320 < LDS_DW

// ---------- async global -> LDS staging (CDNA5 path), with sync fallback ----------
#if __has_builtin(__builtin_amdgcn_global_load_async_to_lds_b64)
#define USE_ASYNC 1
#if __has_builtin(__builtin_amdgcn_s_wait_asynccnt)
#define WAIT_ASYNC(n) __builtin_amdgcn_s_wait_asynccnt(n)
#else
#define WAIT_ASYNC(n) asm volatile("s_wait_asynccnt %0" :: "i"(n) : "memory")
#endif
#else
#define USE_ASYNC 0
#define WAIT_ASYNC(n) do {} while (0)
#endif

typedef __attribute__((address_space(1))) v2i* gptr_v2i;
typedef __attribute__((address_space(3))) v2i* lptr_v2i;

__device__ __forceinline__ void stage_b64(const float2* __restrict__ g, float* lp)
{
#if USE_ASYNC
    __builtin_amdgcn_global_load_async_to_lds_b64(
        (gptr_v2i)g, (lptr_v2i)lp, 0, 0);
#else
    const float2 v = *g;
    lp[0] = v.x;
    lp[1] = v.y;
#endif
}

__device__ __forceinline__ void issue_chunk(const float2* __restrict__ z,
                                            const float2* __restrict__ filt,
                                            float* __restrict__ lds,
                                            int ch, int buf, int pix0,
                                            int tr, int pl)
{
    float* zb = lds + buf * BUFSZ;
    float* fb = zb + ZSZ;
    const int c0 = ch * CT;
    // z chunk: 32 (b,c) rows x 16 pixels -> 2 async ops / thread
    #pragma unroll
    for (int r = 0; r < 2; ++r) {
        const int row = tr + r * 16;            // 0..31
        const int b   = row >> 2;
        const int cc  = row & 3;
        stage_b64(&z[(b * CIN + c0 + cc) * HW + pix0 + pl],
                  zb + cc * KZ + b * RS + pl * 2);
    }
    // filt chunk: 128 (c,p) rows x 16 pixels -> 8 async ops / thread
    #pragma unroll
    for (int r = 0; r < 8; ++r) {
        const int row = tr + r * 16;            // 0..127
        const int cc  = row >> 5;
        const int p   = row & 31;
        stage_b64(&filt[((c0 + cc) * COUT + p) * HW + pix0 + pl],
                  fb + cc * KS + p * RS + pl * 2);
    }
}

__global__ __launch_bounds__(256)
void fftconv_cgemm_wmma(const float2* __restrict__ z,
                        const float2* __restrict__ filt,
                        const float2* __restrict__ bias,
                        float2* __restrict__ out)
{
    __shared__ float lds[LDS_DW];

    const int tid  = threadIdx.x;
    const int pix0 = blockIdx.x * PIX;     // linear pixel = h*W + w; tiles stay in-row
    const int lane = tid & 31;
    const int wv   = tid >> 5;             // wave 0..7
    const int pl   = tid & 15;             // pixel lane for staging
    const int tr   = tid >> 4;             // thread-row 0..15 for staging

    // Accumulators: P = A*Fr, Q = A*Fi; A rows 0-7 = Zr(b), rows 8-15 = Zi(b)
    v8f P[2][2] = {};
    v8f Q[2][2] = {};

    const int m    = lane & 15;            // M for A-tile, N for B-tile
    const int koff = (lane >> 4) << 1;     // lanes 16-31 hold K+2

    issue_chunk(z, filt, lds, 0, 0, pix0, tr, pl);   // prologue: chunk 0 in flight

    for (int ch = 0; ch < NCHUNK; ++ch) {
        const int buf = ch & 1;
        if (ch + 1 < NCHUNK) {
            issue_chunk(z, filt, lds, ch + 1, buf ^ 1, pix0, tr, pl);
            WAIT_ASYNC(10);                // chunk ch landed (in-order), ch+1 may fly
        } else {
            WAIT_ASYNC(0);
        }
        __syncthreads();

        const float* zb = lds + buf * BUFSZ;
        const float* fb = zb + ZSZ;
        #pragma unroll
        for (int px = 0; px < 2; ++px) {
            const int pix = wv * 2 + px;
            v2f a;                          // A 16x4: [Zr(b) ; Zi(b)], K = koff+j
            const int za = koff * KZ + (m & 7) * RS + pix * 2 + (m >> 3);
            a.x = zb[za];
            a.y = zb[za + KZ];
            #pragma unroll
            for (int nt = 0; nt < 2; ++nt) {
                const int p  = nt * 16 + m;
                const int fa = koff * KS + p * RS + pix * 2;
                v2f br, bi;                 // B 4x16, K = koff+j
                br.x = fb[fa];
                br.y = fb[fa + KS];
                bi.x = fb[fa + 1];
                bi.y = fb[fa + KS + 1];
                P[px][nt] = __builtin_amdgcn_wmma_f32_16x16x4_f32(
                    false, a, false, br, (short)0, P[px][nt], false, false);
                Q[px][nt] = __builtin_amdgcn_wmma_f32_16x16x4_f32(
                    false, a, false, bi, (short)0, Q[px][nt], false, false);
            }
        }
        __syncthreads();                    // buf reused two iterations later
    }

    // ---------- combine complex halves + transpose through LDS ----------
    // P lanes<16 = Zr*Fr, lanes>=16 = Zi*Fr ; Q lanes<16 = Zr*Fi, lanes>=16 = Zi*Fi
    // Re = P - swap16(Q) (valid lanes 0-15), Im = P + swap16(Q) (valid lanes 16-31)
    #pragma unroll
    for (int px = 0; px < 2; ++px) {
        const int pix = wv * 2 + px;
        #pragma unroll
        for (int nt = 0; nt < 2; ++nt) {
            #pragma unroll
            for (int v = 0; v < 8; ++v) {
                const float pv = P[px][nt][v];
                const float qs = __shfl_xor(Q[px][nt][v], 16, 32);
                const float U  = pv - qs;   // Out.re, lanes 0-15 (b=v, n=lane)
                const float V  = pv + qs;   // Out.im, lanes 16-31 (b=v, n=lane-16)
                const int  n   = lane & 15;
                if (lane < 16)
                    lds[OLR + pix * OP + v * 32 + nt * 16 + n] = U;
                else
                    lds[OLI + pix * OP + v * 32 + nt * 16 + n] = V;
            }
        }
    }
    __syncthreads();

    // ---------- bias reduction (coalesced stream) + coalesced store ----------
    const int pixE = tid & 15;
    const int prow = tid >> 4;
    #pragma unroll
    for (int half = 0; half < 2; ++half) {
        const int p = prow + half * 16;
        float ar = 0.f, ai = 0.f;
        #pragma unroll 8
        for (int c = 0; c < CIN; ++c) {
            const float2 bv = bias[(c * COUT + p) * HW + pix0 + pixE];
            ar += bv.x; ai += bv.y;
        }
        #pragma unroll
        for (int b = 0; b < BATCH; ++b) {
            float2 o;
            o.x = lds[OLR + pixE * OP + b * 32 + p] + ar;
            o.y = lds[OLI + pixE * OP + b * 32 + p] + ai;
            out[(b * COUT + p) * HW + pix0 + pixE] = o;
        }
    }
}

extern "C" void kernel_launch(void* const* d_in, const int* in_sizes, int n_in,
                              void* d_out, int out_size, void* d_ws, size_t ws_size,
                              hipStream_t stream)
{
    const float2* z    = (const float2*)d_in[0];  // complex64 -> interleaved float2
    const float2* filt = (const float2*)d_in[1];
    const float2* bias = (const float2*)d_in[2];
    float2*       out  = (float2*)d_out;

    dim3 grid(HW / PIX);   // 1024 workgroups
    dim3 block(256);       // 8 wave32s
    fftconv_cgemm_wmma<<<grid, block, 0, stream>>>(z, filt, bias, out);
}